// ChunkedKasminaLayer_43104291782823
// MI455X (gfx1250) — compile-verified
//
#include <hip/hip_runtime.h>

typedef __attribute__((ext_vector_type(2))) float v2f;
typedef __attribute__((ext_vector_type(8))) float v8f;

#define B_SZ   8192
#define D_IN   1024
#define D_OUT  4096
#define NSEED  64
#define CCH    64      // chunk size C per seed
#define HDIM   128     // blueprint hidden

#define MT     128     // rows per workgroup
#define KC     64      // K chunk for base GEMM
#define NKC    (D_IN / KC)
#define PX     68      // LDS pitch (floats) for 64-wide tiles; 68*4=272=16*17 (b128-aligned, bank stride 4)
#define PH     132     // LDS pitch (floats) for 128-wide tiles; 132*4=528=16*33

// LDS arena (floats):
//   [0 .. 2*MT*PX)            : x double buffer (stage 1) -> reused as h tile [MT][PH] (fits: MT*PH=16896 <= 17408)
//   [2*MT*PX .. +2*CCH*PX)    : Wb double buffer -> reused as W1 [HDIM][PX] (=8704, exact) / W2 [CCH][PH] (=8448)
//   [.. +MT*PX)               : y tile, persistent through the whole kernel
#define SMEM_FLOATS (2 * MT * PX + 2 * CCH * PX + MT * PX)

__device__ __forceinline__ void async_b128(const float* gbase, uint32_t gbyteoff, const float* ldsp) {
    uint32_t ldsb = (uint32_t)(uintptr_t)ldsp;   // flat->LDS aperture: low 32 bits are the LDS byte offset
    asm volatile("global_load_async_to_lds_b128 %0, %1, %2 offset:0"
                 :: "v"(ldsb), "v"(gbyteoff), "s"(gbase)
                 : "memory");
}
__device__ __forceinline__ void wait_async_le(int) {}
__device__ __forceinline__ void wait_async_12() { asm volatile("s_wait_asynccnt 0xc" ::: "memory"); }
__device__ __forceinline__ void wait_async_0()  { asm volatile("s_wait_asynccnt 0x0" ::: "memory"); }

// Issue one K-chunk of stage-1 operands: x [128 x 64] (8 x b128/thread) + Wb [64 x 64] (4 x b128/thread)
__device__ __forceinline__ void issue_chunk(const float* __restrict__ x, const float* __restrict__ Wb,
                                            int rowBase, int colBase, int kc0,
                                            float* bufX, float* bufW, int tid)
{
    #pragma unroll
    for (int i = 0; i < 8; ++i) {
        int e = (i * 256 + tid) * 4;                 // 0..8191
        int r = e >> 6, c = e & 63;
        async_b128(x, (uint32_t)(((rowBase + r) * D_IN + kc0 + c) * 4), &bufX[r * PX + c]);
    }
    #pragma unroll
    for (int i = 0; i < 4; ++i) {
        int e = (i * 256 + tid) * 4;                 // 0..4095
        int n = e >> 6, k = e & 63;
        async_b128(Wb, (uint32_t)(((colBase + n) * D_IN + kc0 + k) * 4), &bufW[n * PX + k]);
    }
}

__global__ __launch_bounds__(256)
void kasmina_fused(const float* __restrict__ x,
                   const float* __restrict__ Wb,
                   const float* __restrict__ bb,
                   const float* __restrict__ W1,
                   const float* __restrict__ b1,
                   const float* __restrict__ W2,
                   const float* __restrict__ b2,
                   const float* __restrict__ alpha,
                   const int*   __restrict__ active,
                   float* __restrict__ out)
{
    __shared__ __align__(16) float smem[SMEM_FLOATS];
    float* const sXbuf = smem;                       // 2 x [MT*PX]
    float* const sWbuf = smem + 2 * MT * PX;         // 2 x [CCH*PX]
    float* const sY    = smem + 2 * MT * PX + 2 * CCH * PX;  // [MT*PX]
    float* const sH    = sXbuf;                      // stage 2+: [MT][PH]
    float* const sW    = sWbuf;                      // stage 2+: W1 [HDIM][PX] / W2 [CCH][PH]

    const int s       = blockIdx.x;                  // seed index
    const int rowBase = blockIdx.y * MT;             // batch rows
    const int colBase = s * CCH;                     // this seed's output column chunk

    const int tid  = threadIdx.x;
    const int wave = tid >> 5;
    const int lane = tid & 31;
    const int lr   = lane & 15;                      // row-in-tile (A) / col-in-tile (B, C/D)
    const int hi   = lane >> 4;                      // lane half: selects K pair / M+8
    const int m0   = wave * 16;                      // this wave's row strip

    // Warm L2/WGP$ for the stage-2/3 weights while stage 1 streams (global_prefetch_b8).
    __builtin_prefetch(&W1[((size_t)s * CCH) * HDIM + tid * 32], 0, 1);
    __builtin_prefetch(&W2[((size_t)s * HDIM) * CCH + tid * 32], 0, 1);

    // ============ Stage 1: y = x[rows,:] @ Wb[cols,:]^T + bb ============
    // Double-buffered async global->LDS pipeline: chunk k+1 in flight while chunk k runs on WMMA.
    v8f acc[4] = {};

    issue_chunk(x, Wb, rowBase, colBase, 0, sXbuf, sWbuf, tid);

    for (int kc = 0; kc < NKC; ++kc) {
        const int cur = kc & 1;
        float* const bX = sXbuf + cur * (MT * PX);
        float* const bW = sWbuf + cur * (CCH * PX);

        if (kc + 1 < NKC) {
            issue_chunk(x, Wb, rowBase, colBase, (kc + 1) * KC,
                        sXbuf + (cur ^ 1) * (MT * PX), sWbuf + (cur ^ 1) * (CCH * PX), tid);
            wait_async_12();                         // chunk kc's 12 asyncs retired (in-order)
        } else {
            wait_async_0();
        }
        __syncthreads();                             // chunk kc visible to all waves

        #pragma unroll 4
        for (int k0 = 0; k0 < KC; k0 += 4) {
            // A fragment 16x4 f32: lanes 0-15 -> K=k0,k0+1 ; lanes 16-31 -> K=k0+2,k0+3
            v2f a = *reinterpret_cast<const v2f*>(&bX[(m0 + lr) * PX + k0 + 2 * hi]);
            #pragma unroll
            for (int t = 0; t < 4; ++t) {
                v2f b = *reinterpret_cast<const v2f*>(&bW[(t * 16 + lr) * PX + k0 + 2 * hi]);
                acc[t] = __builtin_amdgcn_wmma_f32_16x16x4_f32(
                    false, a, false, b, (short)0, acc[t], false, false);
            }
        }
        __syncthreads();                             // done reading before buffer is re-filled
    }

    // bias add; park y tile in LDS (stage-2 A operand + final blend input)
    #pragma unroll
    for (int t = 0; t < 4; ++t) {
        float bv = bb[colBase + t * 16 + lr];        // N = lr for every acc VGPR
        #pragma unroll
        for (int r = 0; r < 8; ++r) {
            sY[(m0 + r + 8 * hi) * PX + (t * 16 + lr)] = acc[t][r] + bv;
        }
    }
    // stage W1[s] as B-matrix [n(128)][k(64)]
    #pragma unroll
    for (int i = 0; i < 32; ++i) {
        int e = i * 256 + tid;                       // 0..8191
        int k = e >> 7, n = e & 127;
        sW[n * PX + k] = W1[((size_t)s * CCH + k) * HDIM + n];
    }
    __syncthreads();

    // ============ Stage 2: h = relu(y @ W1[s] + b1[s]) ============
    v8f hacc[8] = {};
    #pragma unroll 4
    for (int k0 = 0; k0 < CCH; k0 += 4) {
        v2f a = *reinterpret_cast<const v2f*>(&sY[(m0 + lr) * PX + k0 + 2 * hi]);
        #pragma unroll
        for (int t = 0; t < 8; ++t) {
            v2f b = *reinterpret_cast<const v2f*>(&sW[(t * 16 + lr) * PX + k0 + 2 * hi]);
            hacc[t] = __builtin_amdgcn_wmma_f32_16x16x4_f32(
                false, a, false, b, (short)0, hacc[t], false, false);
        }
    }
    #pragma unroll
    for (int t = 0; t < 8; ++t) {
        float bv = b1[(size_t)s * HDIM + t * 16 + lr];
        #pragma unroll
        for (int r = 0; r < 8; ++r) {
            float hv = hacc[t][r] + bv;
            sH[(m0 + r + 8 * hi) * PH + (t * 16 + lr)] = hv > 0.f ? hv : 0.f;
        }
    }
    __syncthreads();                                 // stage-2 sW reads done; h fully stored
    // stage W2[s] as B-matrix [n(64)][k(128)]
    #pragma unroll
    for (int i = 0; i < 32; ++i) {
        int e = i * 256 + tid;                       // 0..8191
        int k = e >> 6, n = e & 63;
        sW[n * PH + k] = W2[((size_t)s * HDIM + k) * CCH + n];
    }
    __syncthreads();

    // ============ Stage 3: bp = h @ W2[s] + b2[s]; alpha-blend; store ============
    v8f pacc[4] = {};
    #pragma unroll 4
    for (int k0 = 0; k0 < HDIM; k0 += 4) {
        v2f a = *reinterpret_cast<const v2f*>(&sH[(m0 + lr) * PH + k0 + 2 * hi]);
        #pragma unroll
        for (int t = 0; t < 4; ++t) {
            v2f b = *reinterpret_cast<const v2f*>(&sW[(t * 16 + lr) * PH + k0 + 2 * hi]);
            pacc[t] = __builtin_amdgcn_wmma_f32_16x16x4_f32(
                false, a, false, b, (short)0, pacc[t], false, false);
        }
    }

    const float aval = alpha[s] * (active[s] != 0 ? 1.f : 0.f);
    #pragma unroll
    for (int t = 0; t < 4; ++t) {
        float bv = b2[(size_t)s * CCH + t * 16 + lr];
        #pragma unroll
        for (int r = 0; r < 8; ++r) {
            int m = m0 + r + 8 * hi;
            int n = t * 16 + lr;
            float yv = sY[m * PX + n];
            float bp = pacc[t][r] + bv;
            out[(size_t)(rowBase + m) * D_OUT + colBase + n] =
                (1.f - aval) * yv + aval * bp;
        }
    }
}

extern "C" void kernel_launch(void* const* d_in, const int* in_sizes, int n_in,
                              void* d_out, int out_size, void* d_ws, size_t ws_size,
                              hipStream_t stream) {
    const float* x     = (const float*)d_in[0];
    const float* Wb    = (const float*)d_in[1];
    const float* bb    = (const float*)d_in[2];
    const float* W1    = (const float*)d_in[3];
    const float* b1    = (const float*)d_in[4];
    const float* W2    = (const float*)d_in[5];
    const float* b2    = (const float*)d_in[6];
    const float* alpha = (const float*)d_in[7];
    const int*   act   = (const int*)d_in[8];
    float* out = (float*)d_out;

    dim3 grid(NSEED, B_SZ / MT);   // 64 seeds x 64 row-blocks = 4096 WGs
    kasmina_fused<<<grid, dim3(256), 0, stream>>>(x, Wb, bb, W1, b1, W2, b2, alpha, act, out);
}